// DeformAtten1D_34454227649274
// MI455X (gfx1250) — compile-verified
//
#include <hip/hip_runtime.h>
#include <math.h>

#define Bsz 8
#define Lsz 1024
#define Csz 512
#define Hh  8
#define Gg  4
#define Kk  7
#define GD  128
#define HD  64

#define PADK 20   // 16 k-floats + 4 pad: stride-20 word layout is bank-conflict-free
#define PADM 72   // 64 m-floats + 8 pad: stride-72 rows, halves hit disjoint banks

typedef __attribute__((ext_vector_type(2))) float v2f;
typedef __attribute__((ext_vector_type(8))) float v8f;

#if __has_builtin(__builtin_amdgcn_sched_barrier)
#define SCHED_FENCE() __builtin_amdgcn_sched_barrier(0)
#else
#define SCHED_FENCE()
#endif

__device__ __forceinline__ v8f wmma_f32(v2f a, v2f b, v8f c) {
  // D = A(16x4) * B(4x16) + C, fp32, RNE — exact-precision matrix pipe
  return __builtin_amdgcn_wmma_f32_16x16x4_f32(false, a, false, b, (short)0, c, false, false);
}

// ---------------------------------------------------------------------------
// Kernel 1: projection GEMM  out[b,o,l] = sum_c w[o,c] x[b,l,c] + bias[o] (+rpb)
// Block = 128 thr (4 waves), block tile M=64(o) x N=64(l), K-tile 16, LDS
// double-buffered; prefetch pinned before compute; fragments register-batched.
// ---------------------------------------------------------------------------
__global__ __launch_bounds__(128) void proj_kernel(
    const float* __restrict__ x, const float* __restrict__ w,
    const float* __restrict__ bias, const float* __restrict__ rpb,
    float* __restrict__ out) {
  __shared__ __align__(16) float AS[2][64 * PADK];
  __shared__ __align__(16) float BS[2][64 * PADK];
  const int tid  = threadIdx.x;
  const int lane = tid & 31, wv = tid >> 5;
  const int lm = lane & 15, hl = lane >> 4;
  const int n0 = blockIdx.x * 64;
  const int m0 = blockIdx.y * 64;
  const int b  = blockIdx.z;
  const float* xb = x + (size_t)b * Lsz * Csz;

  // tile fetch: 64 rows x 16 floats = 1024 floats; 128 thr * 2 float4 each
  const int r0 = tid >> 2, c0 = (tid & 3) * 4;
  const int r1 = r0 + 32;
  float4 ra0, ra1, rb0, rb1;
  auto fetch = [&](int kk) {
    ra0 = *(const float4*)(w  + (size_t)(m0 + r0) * Csz + kk + c0);
    ra1 = *(const float4*)(w  + (size_t)(m0 + r1) * Csz + kk + c0);
    rb0 = *(const float4*)(xb + (size_t)(n0 + r0) * Csz + kk + c0);
    rb1 = *(const float4*)(xb + (size_t)(n0 + r1) * Csz + kk + c0);
  };
  auto stage = [&](int buf) {
    *(float4*)&AS[buf][r0 * PADK + c0] = ra0;
    *(float4*)&AS[buf][r1 * PADK + c0] = ra1;
    *(float4*)&BS[buf][r0 * PADK + c0] = rb0;
    *(float4*)&BS[buf][r1 * PADK + c0] = rb1;
  };

  v8f acc[4] = {};
  fetch(0);
  stage(0);
  __syncthreads();
  const int NT = Csz / 16;
  for (int kt = 0; kt < NT; ++kt) {
    const int cur = kt & 1;
    if (kt + 1 < NT) fetch((kt + 1) * 16);
    SCHED_FENCE();   // keep the global prefetch above the compute

    // register-batch all LDS fragments for this K-tile, then one WMMA burst
    v2f afrag[4], bfrag[4][4];
#pragma unroll
    for (int k4 = 0; k4 < 4; ++k4) {
      const int ka = k4 * 4 + 2 * hl;
      afrag[k4] = *(const v2f*)&AS[cur][(wv * 16 + lm) * PADK + ka];
#pragma unroll
      for (int t = 0; t < 4; ++t)
        bfrag[k4][t] = *(const v2f*)&BS[cur][(t * 16 + lm) * PADK + ka];
    }
#pragma unroll
    for (int k4 = 0; k4 < 4; ++k4)
#pragma unroll
      for (int t = 0; t < 4; ++t)      // 4 independent acc chains interleaved
        acc[t] = wmma_f32(afrag[k4], bfrag[k4][t], acc[t]);

    SCHED_FENCE();
    if (kt + 1 < NT) stage(cur ^ 1);
    __syncthreads();
  }

  float* ob = out + (size_t)b * Csz * Lsz;
#pragma unroll
  for (int t = 0; t < 4; ++t)
#pragma unroll
    for (int r = 0; r < 8; ++r) {
      const int m = m0 + wv * 16 + r + 8 * hl;
      const int n = n0 + t * 16 + lm;
      float val = acc[t][r] + bias[m];
      if (rpb) val += rpb[(size_t)m * Lsz + n];
      ob[(size_t)m * Lsz + n] = val;                 // coalesced in n
    }
}

// ---------------------------------------------------------------------------
// Kernel 2a: collapse conv(GDxGDx7) -> 1x1 into one 128x7 stencil (exact)
// ---------------------------------------------------------------------------
__global__ void weff_kernel(const float* __restrict__ w1, const float* __restrict__ b1,
                            const float* __restrict__ w2, const float* __restrict__ b2,
                            float* __restrict__ weff) {
  const int i = blockIdx.x * blockDim.x + threadIdx.x;
  if (i < GD * Kk) {
    const int ic = i / Kk, t = i % Kk;
    float s = 0.f;
    for (int oc = 0; oc < GD; ++oc) s += w2[oc] * w1[(oc * GD + ic) * Kk + t];
    weff[i] = s;
  } else if (i == GD * Kk) {
    float s = b2[0];
    for (int oc = 0; oc < GD; ++oc) s += w2[oc] * b1[oc];
    weff[i] = s;
  }
}

// ---------------------------------------------------------------------------
// Kernel 2b: off[bg,l] = tanh(beff + sum_{ic,t} Weff[ic,t]*q[b,g*GD+ic,l+t-3]) * 7
// ---------------------------------------------------------------------------
__global__ void off_kernel(const float* __restrict__ q, const float* __restrict__ weff,
                           float* __restrict__ off) {
  const int idx = blockIdx.x * 256 + threadIdx.x;   // B*G*L threads
  const int bg = idx >> 10, l = idx & (Lsz - 1);
  const int b = bg >> 2, g = bg & 3;
  const float* qc = q + (size_t)b * Csz * Lsz + (size_t)(g * GD) * Lsz;
  float s = weff[GD * Kk];
  for (int ic = 0; ic < GD; ++ic) {
    const float* row  = qc + (size_t)ic * Lsz;
    const float* wrow = weff + ic * Kk;
#pragma unroll
    for (int t = 0; t < Kk; ++t) {
      const int ll = l + t - 3;
      const float xv = (ll >= 0 && ll < Lsz) ? row[ll] : 0.f;
      s += xv * wrow[t];
    }
  }
  off[idx] = tanhf(s) * (float)Kk;
}

// ---------------------------------------------------------------------------
// Kernel 3: bilinear resample of k and v along L (zeros padding)
// ---------------------------------------------------------------------------
__global__ void sample_kernel(const float* __restrict__ kin, const float* __restrict__ vin,
                              const float* __restrict__ off,
                              float* __restrict__ ks, float* __restrict__ vs) {
  const size_t idx = (size_t)blockIdx.x * 256 + threadIdx.x;   // B*C*L threads
  const int l = (int)(idx & (Lsz - 1));
  const size_t row = idx >> 10;        // = bg*GD + c  (== b*C + g*GD + c)
  const int bg = (int)(row >> 7);
  const float pos = (float)l + off[(size_t)bg * Lsz + l];
  const float x0 = floorf(pos);
  const float w  = pos - x0;
  const int i0 = (int)x0, i1 = i0 + 1;
  const float* kr = kin + row * Lsz;
  const float* vr = vin + row * Lsz;
  const float k0 = (i0 >= 0 && i0 < Lsz) ? kr[i0] : 0.f;
  const float k1 = (i1 >= 0 && i1 < Lsz) ? kr[i1] : 0.f;
  const float v0 = (i0 >= 0 && i0 < Lsz) ? vr[i0] : 0.f;
  const float v1 = (i1 >= 0 && i1 < Lsz) ? vr[i1] : 0.f;
  ks[idx] = k0 * (1.f - w) + k1 * w;
  vs[idx] = v0 * (1.f - w) + v1 * w;
}

// ---------------------------------------------------------------------------
// Kernel 4: flash-style attention. One wave per (b, h, 16-query tile).
// S = Q^T K via WMMA (K=64, Q tile register-resident), online softmax,
// O += P V^T via WMMA.  K loads batched before the S chain; V tile prefetched
// before the softmax VALU work so exp/shuffle hides its latency.
// ---------------------------------------------------------------------------
__global__ __launch_bounds__(32) void attn_kernel(
    const float* __restrict__ q, const float* __restrict__ ks,
    const float* __restrict__ vs, float* __restrict__ oh) {
  __shared__ float pshm[256];                     // 16x16 P tile
  const int lane = threadIdx.x, lm = lane & 15, hl = lane >> 4;
  const int i0 = blockIdx.x * 16, h = blockIdx.y, b = blockIdx.z;
  const float* qb = q  + ((size_t)b * Csz + h * HD) * Lsz;
  const float* kb = ks + ((size_t)b * Csz + h * HD) * Lsz;
  const float* vb = vs + ((size_t)b * Csz + h * HD) * Lsz;

  // Q tile in A-operand layout: aQ[kk] = A[m=lm][4kk+2hl .. +1]
  v2f aQ[16];
#pragma unroll
  for (int kk = 0; kk < 16; ++kk) {
    const int kr = 4 * kk + 2 * hl;
    aQ[kk].x = qb[(size_t)kr * Lsz + i0 + lm];
    aQ[kk].y = qb[(size_t)(kr + 1) * Lsz + i0 + lm];
  }

  float mrow[8], lrow[8];
#pragma unroll
  for (int r = 0; r < 8; ++r) { mrow[r] = -3.0e38f; lrow[r] = 0.f; }
  v8f oacc[4] = {};

  for (int j0 = 0; j0 < Lsz; j0 += 16) {
    // batch all K-operand loads, then run the 16-deep WMMA accumulation chain
    v2f bk[16];
#pragma unroll
    for (int kk = 0; kk < 16; ++kk) {
      const int kr = 4 * kk + 2 * hl;
      bk[kk].x = kb[(size_t)kr * Lsz + j0 + lm];
      bk[kk].y = kb[(size_t)(kr + 1) * Lsz + j0 + lm];
    }
    v8f s = {};
#pragma unroll
    for (int kk = 0; kk < 16; ++kk) s = wmma_f32(aQ[kk], bk[kk], s);

    // prefetch V tile (independent of S) so softmax VALU hides the loads
    v2f bv16[16];
#pragma unroll
    for (int k4 = 0; k4 < 4; ++k4)
#pragma unroll
      for (int t = 0; t < 4; ++t)
        bv16[k4 * 4 + t] =
            *(const v2f*)(vb + (size_t)(16 * t + lm) * Lsz + j0 + k4 * 4 + 2 * hl);

    // online softmax (rows live in 16-lane halves; xor masks 1..8 stay in-half)
    float sf[8], tm[8];
#pragma unroll
    for (int r = 0; r < 8; ++r) { sf[r] = s[r] * 0.125f; tm[r] = sf[r]; }
#pragma unroll
    for (int m = 1; m < 16; m <<= 1)
#pragma unroll
      for (int r = 0; r < 8; ++r) tm[r] = fmaxf(tm[r], __shfl_xor(tm[r], m, 32));
    float scalef[8], ps[8];
#pragma unroll
    for (int r = 0; r < 8; ++r) {
      const float nm = fmaxf(mrow[r], tm[r]);
      scalef[r] = expf(mrow[r] - nm);
      mrow[r] = nm;
      const float p = expf(sf[r] - nm);
      sf[r] = p;
      ps[r] = p;
    }
#pragma unroll
    for (int m = 1; m < 16; m <<= 1)
#pragma unroll
      for (int r = 0; r < 8; ++r) ps[r] += __shfl_xor(ps[r], m, 32);
#pragma unroll
    for (int r = 0; r < 8; ++r) lrow[r] = lrow[r] * scalef[r] + ps[r];
#pragma unroll
    for (int t = 0; t < 4; ++t)
#pragma unroll
      for (int r = 0; r < 8; ++r) oacc[t][r] = oacc[t][r] * scalef[r];

    // stage P (D layout -> LDS -> A layout)
#pragma unroll
    for (int r = 0; r < 8; ++r) pshm[(r + 8 * hl) * 16 + lm] = sf[r];
    __syncthreads();
#pragma unroll
    for (int k4 = 0; k4 < 4; ++k4) {
      v2f a = *(const v2f*)&pshm[lm * 16 + k4 * 4 + 2 * hl];
#pragma unroll
      for (int t = 0; t < 4; ++t) oacc[t] = wmma_f32(a, bv16[k4 * 4 + t], oacc[t]);
    }
    __syncthreads();
  }
#pragma unroll
  for (int t = 0; t < 4; ++t)
#pragma unroll
    for (int r = 0; r < 8; ++r) {
      const float val = oacc[t][r] / lrow[r];
      oh[((size_t)b * Csz + h * HD + 16 * t + lm) * Lsz + i0 + r + 8 * hl] = val;
    }
}

// ---------------------------------------------------------------------------
// Kernel 5: output projection  out[b,l,o] = sum_c oh[b,c,l]*w_out[o,c] + b_out[o]
// Block = 128 thr (4 waves), tile M=64(l) x N=64(o), K-tile 16, LDS staged.
// A (= oh) is k-major in memory -> stored transposed in LDS [k][PADM].
// ---------------------------------------------------------------------------
__global__ __launch_bounds__(128) void outproj_kernel(
    const float* __restrict__ oh, const float* __restrict__ wout,
    const float* __restrict__ bout, float* __restrict__ out) {
  __shared__ __align__(16) float AS[2][16 * PADM];
  __shared__ __align__(16) float BS[2][64 * PADK];
  const int tid  = threadIdx.x;
  const int lane = tid & 31, wv = tid >> 5;
  const int lm = lane & 15, hl = lane >> 4;
  const int n0 = blockIdx.x * 64;
  const int m0 = blockIdx.y * 64;
  const int b  = blockIdx.z;
  const float* ob = oh + (size_t)b * Csz * Lsz;

  // A tile: 16 k-rows x 64 m floats; B tile: 64 n-rows x 16 k floats
  const int ra_row = tid >> 4, ra_col = (tid & 15) * 4;   // +128 -> row+8
  const int rb_row = tid >> 2, rb_col = (tid & 3) * 4;    // +128 -> row+32
  float4 fa0, fa1, fb0, fb1;
  auto fetch = [&](int kk) {
    fa0 = *(const float4*)(ob   + (size_t)(kk + ra_row) * Lsz + m0 + ra_col);
    fa1 = *(const float4*)(ob   + (size_t)(kk + ra_row + 8) * Lsz + m0 + ra_col);
    fb0 = *(const float4*)(wout + (size_t)(n0 + rb_row) * Csz + kk + rb_col);
    fb1 = *(const float4*)(wout + (size_t)(n0 + rb_row + 32) * Csz + kk + rb_col);
  };
  auto stage = [&](int buf) {
    *(float4*)&AS[buf][ra_row * PADM + ra_col]        = fa0;
    *(float4*)&AS[buf][(ra_row + 8) * PADM + ra_col]  = fa1;
    *(float4*)&BS[buf][rb_row * PADK + rb_col]        = fb0;
    *(float4*)&BS[buf][(rb_row + 32) * PADK + rb_col] = fb1;
  };

  v8f acc[4] = {};
  fetch(0);
  stage(0);
  __syncthreads();
  const int NT = Csz / 16;
  for (int kt = 0; kt < NT; ++kt) {
    const int cur = kt & 1;
    if (kt + 1 < NT) fetch((kt + 1) * 16);
    SCHED_FENCE();   // keep the global prefetch above the compute

    v2f afrag[4], bfrag[4][4];
#pragma unroll
    for (int k4 = 0; k4 < 4; ++k4) {
      const int ka = k4 * 4 + 2 * hl;
      afrag[k4].x = AS[cur][ka * PADM + wv * 16 + lm];
      afrag[k4].y = AS[cur][(ka + 1) * PADM + wv * 16 + lm];
#pragma unroll
      for (int t = 0; t < 4; ++t)
        bfrag[k4][t] = *(const v2f*)&BS[cur][(t * 16 + lm) * PADK + ka];
    }
#pragma unroll
    for (int k4 = 0; k4 < 4; ++k4)
#pragma unroll
      for (int t = 0; t < 4; ++t)
        acc[t] = wmma_f32(afrag[k4], bfrag[k4][t], acc[t]);

    SCHED_FENCE();
    if (kt + 1 < NT) stage(cur ^ 1);
    __syncthreads();
  }

  float* dst = out + (size_t)b * Lsz * Csz;
#pragma unroll
  for (int t = 0; t < 4; ++t)
#pragma unroll
    for (int r = 0; r < 8; ++r) {
      const int l = m0 + wv * 16 + r + 8 * hl;
      const int o = n0 + t * 16 + lm;
      dst[(size_t)l * Csz + o] = acc[t][r] + bout[o];   // coalesced in o
    }
}

// ---------------------------------------------------------------------------
extern "C" void kernel_launch(void* const* d_in, const int* in_sizes, int n_in,
                              void* d_out, int out_size, void* d_ws, size_t ws_size,
                              hipStream_t stream) {
  const float* x   = (const float*)d_in[0];
  const float* wq  = (const float*)d_in[1];
  const float* bq  = (const float*)d_in[2];
  const float* wk  = (const float*)d_in[3];
  const float* bk  = (const float*)d_in[4];
  const float* wv  = (const float*)d_in[5];
  const float* bv  = (const float*)d_in[6];
  const float* w1  = (const float*)d_in[7];
  const float* b1  = (const float*)d_in[8];
  const float* w2  = (const float*)d_in[9];
  const float* b2  = (const float*)d_in[10];
  const float* wo  = (const float*)d_in[11];
  const float* bo  = (const float*)d_in[12];
  const float* rpb = (const float*)d_in[13];
  float* out = (float*)d_out;

  const size_t NCL = (size_t)Bsz * Csz * Lsz;   // 4M floats
  float* ws   = (float*)d_ws;
  float* q    = ws;
  float* k    = q + NCL;
  float* v    = k + NCL;
  float* ksmp = v + NCL;
  float* vsmp = ksmp + NCL;
  float* ohb  = vsmp + NCL;
  float* off  = ohb + NCL;
  float* weff = off + (size_t)Bsz * Gg * Lsz;

  dim3 gproj(Lsz / 64, Csz / 64, Bsz);
  proj_kernel<<<gproj, 128, 0, stream>>>(x, wq, bq, nullptr, q);
  proj_kernel<<<gproj, 128, 0, stream>>>(x, wk, bk, nullptr, k);
  proj_kernel<<<gproj, 128, 0, stream>>>(x, wv, bv, rpb, v);

  weff_kernel<<<4, 256, 0, stream>>>(w1, b1, w2, b2, weff);
  off_kernel<<<(Bsz * Gg * Lsz) / 256, 256, 0, stream>>>(q, weff, off);
  sample_kernel<<<(int)(NCL / 256), 256, 0, stream>>>(k, v, off, ksmp, vsmp);

  attn_kernel<<<dim3(Lsz / 16, Hh, Bsz), 32, 0, stream>>>(q, ksmp, vsmp, ohb);
  outproj_kernel<<<dim3(Csz / 64, Lsz / 64, Bsz), 128, 0, stream>>>(ohb, wo, bo, out);
}